// LoRAQKV_66872640799074
// MI455X (gfx1250) — compile-verified
//
#include <hip/hip_runtime.h>
#include <stdint.h>

typedef __attribute__((ext_vector_type(2))) float v2f;
typedef __attribute__((ext_vector_type(8))) float v8f;

// Problem constants (from reference): B=32, N=1024, D=1024, RANK=16
static constexpr int M_TOT  = 32 * 1024;   // 32768 rows (B*N)
static constexpr int K_DIM  = 1024;        // D
static constexpr int N_OUT  = 3 * 1024;    // 3072 output features
static constexpr int LRANK  = 16;
static constexpr float SCALING = 16.0f / 16.0f;  // alpha / rank = 1.0

static constexpr int KB   = 32;            // K chunk staged in LDS
static constexpr int LSTR = 36;            // padded LDS row stride (floats): 16B rows, conflict-free b64 frags
static constexpr int TSTR = 20;            // padded stride for rank-16 t buffer

static constexpr int X_OFF = 0;                 // 128 x LSTR  (x tile)
static constexpr int W_OFF = 128 * LSTR;        //  64 x LSTR  (W tile)
static constexpr int A_OFF = (128 + 64) * LSTR; //  16 x LSTR  (LoRA A tile)
static constexpr int BUF_SZ = (128 + 64 + 16) * LSTR;       // floats per buffer
static constexpr uint32_t BUF_BYTES = BUF_SZ * 4;           // 29952 B

#define WMMA_F32(a, b, c) \
    __builtin_amdgcn_wmma_f32_16x16x4_f32(false, (a), false, (b), (short)0, (c), false, false)

// CDNA5 async DMA: global -> LDS, 16 bytes per lane, tracked by ASYNCcnt.
__device__ __forceinline__ void async_copy_b128(uint32_t lds_byte_off, const float* g) {
    asm volatile("global_load_async_to_lds_b128 %0, %1, off"
                 :: "v"(lds_byte_off), "v"((uint64_t)(uintptr_t)g)
                 : "memory");
}
__device__ __forceinline__ void wait_async0() {
    asm volatile("s_wait_asynccnt 0x0" ::: "memory");
}

__global__ __launch_bounds__(256) void lora_qkv_fused(
    const float* __restrict__ x,    // [M_TOT, K_DIM]
    const float* __restrict__ W,    // [N_OUT, K_DIM]
    const float* __restrict__ bias, // [N_OUT]
    const float* __restrict__ A_q,  // [16, K_DIM]
    const float* __restrict__ B_q,  // [K_DIM, 16]
    const float* __restrict__ A_v,  // [16, K_DIM]
    const float* __restrict__ B_v,  // [K_DIM, 16]
    float* __restrict__ out)        // [M_TOT, N_OUT]
{
    __shared__ __align__(16) float lds[2 * BUF_SZ];
    // generic->LDS-offset: low 32 bits of the flat address are the LDS byte offset
    const uint32_t lds0 = (uint32_t)(uintptr_t)(void*)&lds[0];

    const int tid  = threadIdx.x;
    const int lane = tid & 31;
    const int w    = tid >> 5;     // wave id 0..7
    const int wr   = w >> 1;       // wave row 0..3  (32 rows each)
    const int wc   = w & 1;        // wave col 0..1  (32 cols each)
    const int half = lane >> 4;    // 0 | 1
    const int l16  = lane & 15;

    const int n0 = blockIdx.x * 64;
    const int m0 = blockIdx.y * 128;

    const bool isQ = (n0 < K_DIM);
    const bool isV = (n0 >= 2 * K_DIM);
    const bool need_lora = isQ || isV;           // block-uniform
    const float* Al = isQ ? A_q : A_v;           // always a valid pointer
    const float* Bl = isQ ? B_q : B_v;

    // Stage one K-chunk [kt, kt+KB) into buffer `buf` via async DMA.
    auto stage = [&](int buf, int kt) {
        const uint32_t base = lds0 + (uint32_t)buf * BUF_BYTES;
        #pragma unroll
        for (int it = 0; it < 4; ++it) {           // x: 128 x 32 floats
            int i = tid + it * 256;
            int row = i >> 3, c4 = (i & 7) << 2;
            async_copy_b128(base + (uint32_t)(row * LSTR + c4) * 4u,
                            &x[(size_t)(m0 + row) * K_DIM + kt + c4]);
        }
        #pragma unroll
        for (int it = 0; it < 2; ++it) {           // W: 64 x 32 floats
            int i = tid + it * 256;
            int row = i >> 3, c4 = (i & 7) << 2;
            async_copy_b128(base + (uint32_t)(W_OFF + row * LSTR + c4) * 4u,
                            &W[(size_t)(n0 + row) * K_DIM + kt + c4]);
        }
        if (tid < 128) {                           // LoRA A: 16 x 32 floats
            int row = tid >> 3, c4 = (tid & 7) << 2;
            async_copy_b128(base + (uint32_t)(A_OFF + row * LSTR + c4) * 4u,
                            &Al[(size_t)row * K_DIM + kt + c4]);
        }
    };

    v8f acc[2][2];   // [col-group][row-group] of 16x16 f32 tiles
    v8f accL[2];     // t = x * Al^T accumulators (rank-16), per row-group
    #pragma unroll
    for (int cg = 0; cg < 2; ++cg)
        #pragma unroll
        for (int rg = 0; rg < 2; ++rg) acc[cg][rg] = (v8f){};
    accL[0] = (v8f){};
    accL[1] = (v8f){};

    // ---- software pipeline: prefetch chunk k+1 while computing chunk k ----
    stage(0, 0);
    wait_async0();
    __syncthreads();

    for (int kt = 0; kt < K_DIM; kt += KB) {
        const int cur = (kt / KB) & 1;
        if (kt + KB < K_DIM) stage(cur ^ 1, kt + KB);

        const float* xb = &lds[(size_t)cur * BUF_SZ];
        const float* wb = xb + W_OFF;
        const float* ab = xb + A_OFF;

        #pragma unroll
        for (int kk = 0; kk < KB; kk += 4) {
            const int kb = kk + 2 * half;
            // A fragments (16x4 f32): lane row = l16, K pair selected by half
            v2f a0 = *(const v2f*)&xb[(wr * 32 +      l16) * LSTR + kb];
            v2f a1 = *(const v2f*)&xb[(wr * 32 + 16 + l16) * LSTR + kb];
            // B fragments (4x16 f32) from W rows (NT gemm)
            v2f b0 = *(const v2f*)&wb[(wc * 32 +      l16) * LSTR + kb];
            v2f b1 = *(const v2f*)&wb[(wc * 32 + 16 + l16) * LSTR + kb];
            v2f bl = *(const v2f*)&ab[l16 * LSTR + kb];

            acc[0][0] = WMMA_F32(a0, b0, acc[0][0]);
            acc[0][1] = WMMA_F32(a1, b0, acc[0][1]);
            acc[1][0] = WMMA_F32(a0, b1, acc[1][0]);
            acc[1][1] = WMMA_F32(a1, b1, acc[1][1]);
            // LoRA t accumulation: unconditional (branch-free inner loop)
            accL[0] = WMMA_F32(a0, bl, accL[0]);
            accL[1] = WMMA_F32(a1, bl, accL[1]);
        }

        wait_async0();       // prefetch landed (ds_loads already drained by wmma waits)
        __syncthreads();
    }

    // ---- LoRA epilogue: out_tile += (t * SCALING) @ Bl_slice^T ----
    float* t_lds = &lds[0];  // reuse buffer 0 (128 x TSTR)
    if (need_lora) {
        if (wc == 0) {  // wc=0 and wc=1 waves computed identical t; one writes
            #pragma unroll
            for (int rg = 0; rg < 2; ++rg)
                #pragma unroll
                for (int i = 0; i < 8; ++i) {
                    int row = wr * 32 + rg * 16 + i + 8 * half;  // C-layout row
                    t_lds[row * TSTR + l16] = accL[rg][i] * SCALING;
                }
        }
        __syncthreads();
        #pragma unroll
        for (int cg = 0; cg < 2; ++cg) {
            const int ncol = n0 + wc * 32 + cg * 16 + l16;
            const int nloc = isV ? (ncol - 2 * K_DIM) : ncol;  // column within D
            #pragma unroll
            for (int rg = 0; rg < 2; ++rg) {
                #pragma unroll
                for (int kk = 0; kk < LRANK; kk += 4) {
                    const int kb = kk + 2 * half;
                    v2f a = *(const v2f*)&t_lds[(wr * 32 + rg * 16 + l16) * TSTR + kb];
                    v2f b = *(const v2f*)&Bl[(size_t)nloc * LRANK + kb];
                    acc[cg][rg] = WMMA_F32(a, b, acc[cg][rg]);
                }
            }
        }
    }

    // ---- bias + store ----
    #pragma unroll
    for (int cg = 0; cg < 2; ++cg) {
        const int ncol = n0 + wc * 32 + cg * 16 + l16;
        const float bv = bias[ncol];
        #pragma unroll
        for (int rg = 0; rg < 2; ++rg) {
            const size_t base =
                (size_t)(m0 + wr * 32 + rg * 16 + 8 * half) * N_OUT + ncol;
            #pragma unroll
            for (int i = 0; i < 8; ++i)
                out[base + (size_t)i * N_OUT] = acc[cg][rg][i] + bv;
        }
    }
}

extern "C" void kernel_launch(void* const* d_in, const int* in_sizes, int n_in,
                              void* d_out, int out_size, void* d_ws, size_t ws_size,
                              hipStream_t stream) {
    const float* x   = (const float*)d_in[0];
    const float* W   = (const float*)d_in[1];
    const float* b   = (const float*)d_in[2];
    const float* A_q = (const float*)d_in[3];
    const float* B_q = (const float*)d_in[4];
    const float* A_v = (const float*)d_in[5];
    const float* B_v = (const float*)d_in[6];
    float* out = (float*)d_out;

    dim3 grid(N_OUT / 64, M_TOT / 128);  // 48 x 256 blocks
    lora_qkv_fused<<<grid, 256, 0, stream>>>(x, W, b, A_q, B_q, A_v, B_v, out);
}